// HF_56461640073536
// MI455X (gfx1250) — compile-verified
//
#include <hip/hip_runtime.h>
#include <hip/hip_bf16.h>

typedef __attribute__((ext_vector_type(2))) float v2f;
typedef __attribute__((ext_vector_type(4))) float v4f;
typedef __attribute__((ext_vector_type(8))) float v8f;

#define ZD    128          // feature dim
#define LSTR  132          // padded LDS row stride (floats): conflict-free b64 frag reads
#define WAVES 4            // waves per block (wave32)
#define TROWS 16           // rows per wave tile
#define BROWS (WAVES * TROWS)

// Fused kernel: v_new = v*W^T + b via V_WMMA_F32_16X16X4_F32, then
// z_new = z - 2*v_new*(v_new.z)/||v_new||^2, one HBM pass over z/v/out.
__global__ __launch_bounds__(WAVES * 32)
void hh_wmma_kernel(const float* __restrict__ zg, const float* __restrict__ vg,
                    const float* __restrict__ Wg, const float* __restrict__ bg,
                    float* __restrict__ outg)
{
    extern __shared__ __align__(16) float smem[];
    float* Wl = smem;                        // ZD * LSTR floats  (W, row-major, padded)
    float* zb = Wl + ZD * LSTR;              // WAVES * TROWS * LSTR (z tiles)
    float* vb = zb + WAVES * TROWS * LSTR;   // WAVES * TROWS * LSTR (v tiles / out staging)

    const int tid  = threadIdx.x;
    const int lane = tid & 31;
    const int wave = tid >> 5;
    const int half = lane >> 4;    // 0: lanes 0-15, 1: lanes 16-31
    const int l16  = lane & 15;

    // ---- Stage W (128x128) into padded LDS, fully coalesced b128 loads ----
    for (int i = tid; i < (ZD * ZD) / 4; i += WAVES * 32) {
        int r = (i * 4) / ZD, c = (i * 4) % ZD;
        *(v4f*)(Wl + r * LSTR + c) = *(const v4f*)(Wg + (long)r * ZD + c);
    }

    const long rowBase = (long)blockIdx.x * BROWS + wave * TROWS;
    float* zt = zb + wave * (TROWS * LSTR);
    float* vt = vb + wave * (TROWS * LSTR);

    // ---- Stage this wave's 16x128 z and v tiles, coalesced ----
    for (int i = lane; i < (TROWS * ZD) / 4; i += 32) {
        int r = (i * 4) / ZD, c = (i * 4) % ZD;
        long g = (rowBase + r) * ZD + c;
        *(v4f*)(zt + r * LSTR + c) = *(const v4f*)(zg + g);
        *(v4f*)(vt + r * LSTR + c) = *(const v4f*)(vg + g);
    }
    __syncthreads();

    // ---- Accumulators initialized with bias: C[m][n] = b[n] ----
    // D layout: VGPR g of tile n -> row 8*half+g, col n*16+l16.
    v8f acc[8];
#pragma unroll
    for (int n = 0; n < 8; ++n) {
        float bv = bg[n * 16 + l16];
        acc[n] = (v8f){bv, bv, bv, bv, bv, bv, bv, bv};
    }

    // ---- GEMM: v_new(16x128) = v(16x128) @ W^T, K in steps of 4 ----
    // A frag (16x4):  lane row = l16, K = k0+2*half .. +1          (ds_load_b64)
    // B frag (4x16):  B[k][n] = W[n][k] -> row n*16+l16 of W, same K pair
    for (int k0 = 0; k0 < ZD; k0 += 4) {
        v2f a = *(const v2f*)(vt + l16 * LSTR + k0 + 2 * half);
#pragma unroll
        for (int n = 0; n < 8; ++n) {
            v2f bf = *(const v2f*)(Wl + (n * 16 + l16) * LSTR + k0 + 2 * half);
            acc[n] = __builtin_amdgcn_wmma_f32_16x16x4_f32(
                false, a, false, bf, (short)0, acc[n], false, false);
        }
    }

    // ---- Householder: per-row dot(v_new,z) and ||v_new||^2 ----
    float dotp[8], nsq[8];
#pragma unroll
    for (int g = 0; g < 8; ++g) { dotp[g] = 0.0f; nsq[g] = 0.0f; }
#pragma unroll
    for (int n = 0; n < 8; ++n) {
#pragma unroll
        for (int g = 0; g < 8; ++g) {
            float vn = acc[n][g];
            float zv = zt[(8 * half + g) * LSTR + n * 16 + l16];
            dotp[g] = fmaf(vn, zv, dotp[g]);
            nsq[g]  = fmaf(vn, vn, nsq[g]);
        }
    }

    // Reduce across the 16 lanes that share each row (xor masks stay in-half).
#pragma unroll
    for (int g = 0; g < 8; ++g) {
        float d = dotp[g], s = nsq[g];
#pragma unroll
        for (int m = 1; m < 16; m <<= 1) {
            d += __shfl_xor(d, m, 32);
            s += __shfl_xor(s, m, 32);
        }
        float scale = 2.0f * d / s;
#pragma unroll
        for (int n = 0; n < 8; ++n) {
            int idx = (8 * half + g) * LSTR + n * 16 + l16;
            vt[idx] = fmaf(-scale, acc[n][g], zt[idx]);   // z - scale*v_new
        }
    }
    __syncthreads();

    // ---- Coalesced b128 store of the staged result ----
    for (int i = lane; i < (TROWS * ZD) / 4; i += 32) {
        int r = (i * 4) / ZD, c = (i * 4) % ZD;
        *(v4f*)(outg + (rowBase + r) * ZD + c) = *(const v4f*)(vt + r * LSTR + c);
    }
}

// Scalar fallback for any trailing rows (B % 64) — one wave per row.
__global__ void hh_tail_kernel(const float* __restrict__ zg, const float* __restrict__ vg,
                               const float* __restrict__ Wg, const float* __restrict__ bg,
                               float* __restrict__ outg, long start, long nrows)
{
    long row = start + blockIdx.x;
    if (row >= nrows) return;
    int lane = threadIdx.x;
    float vn[4];
    float dot = 0.0f, nsq = 0.0f;
#pragma unroll
    for (int j = 0; j < 4; ++j) {
        int n = lane * 4 + j;
        float a = bg[n];
        for (int k = 0; k < ZD; ++k)
            a = fmaf(vg[row * ZD + k], Wg[(long)n * ZD + k], a);
        vn[j] = a;
        dot = fmaf(a, zg[row * ZD + n], dot);
        nsq = fmaf(a, a, nsq);
    }
    for (int m = 1; m < 32; m <<= 1) {
        dot += __shfl_xor(dot, m, 32);
        nsq += __shfl_xor(nsq, m, 32);
    }
    float scale = 2.0f * dot / nsq;
#pragma unroll
    for (int j = 0; j < 4; ++j) {
        int n = lane * 4 + j;
        outg[row * ZD + n] = fmaf(-scale, vn[j], zg[row * ZD + n]);
    }
}

extern "C" void kernel_launch(void* const* d_in, const int* in_sizes, int n_in,
                              void* d_out, int out_size, void* d_ws, size_t ws_size,
                              hipStream_t stream)
{
    const float* z = (const float*)d_in[0];
    const float* v = (const float*)d_in[1];
    const float* W = (const float*)d_in[2];
    const float* b = (const float*)d_in[3];
    float* out = (float*)d_out;

    long B = (long)in_sizes[0] / ZD;          // number of rows
    long mainBlocks = B / BROWS;
    long tail = B % BROWS;

    size_t shmem = (size_t)(ZD * LSTR + 2 * WAVES * TROWS * LSTR) * sizeof(float); // 132 KB

    if (mainBlocks > 0)
        hh_wmma_kernel<<<(int)mainBlocks, WAVES * 32, shmem, stream>>>(z, v, W, b, out);
    if (tail > 0)
        hh_tail_kernel<<<(int)tail, 32, 0, stream>>>(z, v, W, b, out, B - tail, B);
}